// Net_inner_53712861003993
// MI455X (gfx1250) — compile-verified
//
#include <hip/hip_runtime.h>

typedef float v2f __attribute__((ext_vector_type(2)));
typedef float v8f __attribute__((ext_vector_type(8)));

#define N_NODES   100000
#define F_IN      256
#define H_DIM     128
#define C_DIM     64
#define N_EDGES   1600000
#define EP_EDGES  500000

// ---------------- utility kernels ----------------

__global__ void fill_zero_kernel(float* __restrict__ p, int n) {
    int i = blockIdx.x * blockDim.x + threadIdx.x;
    if (i < n) p[i] = 0.0f;
}

__global__ void degree_count_kernel(const int* __restrict__ dst,
                                    float* __restrict__ deg, int e) {
    int i = blockIdx.x * blockDim.x + threadIdx.x;
    if (i < e) atomicAdd(&deg[dst[i]], 1.0f);
}

__global__ void deg_to_dis_kernel(float* __restrict__ deg, int n) {
    int i = blockIdx.x * blockDim.x + threadIdx.x;
    if (i < n) deg[i] = rsqrtf(deg[i] + 1.0f);
}

// ---------------- fp32 WMMA GEMM with LDS-staged B panel ----------------
// D[M x NC] = A[M x K] * B[K x NC].
// One block = one 16-row strip of A; wave w owns column tile w (16 cols).
// blockDim.x = (NC/16)*32. M must be a multiple of 16 (100000 = 6250*16).
// B panel of KC rows is staged in LDS cooperatively (coalesced float4), then
// each wave runs a fully unrolled run of V_WMMA_F32_16X16X4_F32 with
// constant-offset ds reads — no per-iteration 64-bit address math.
template <int KC, int NC>
__global__ __launch_bounds__(256) void gemm_wmma_lds_kernel(
    const float* __restrict__ A, const float* __restrict__ B,
    float* __restrict__ D, int K) {
    __shared__ float Bs[KC * NC];

    const int wave = threadIdx.x >> 5;
    const int lane = threadIdx.x & 31;
    const int hi   = lane >> 4;      // 0: lanes 0-15, 1: lanes 16-31
    const int l    = lane & 15;
    const int m0   = blockIdx.x * 16;
    const int n0   = wave * 16;

    v8f c = {};
    // A 16x4 f32 fragment: lanes 0-15 hold (K=k,K=k+1), lanes 16-31 (K=k+2,K=k+3)
    const float* arow = A + (size_t)(m0 + l) * K + 2 * hi;
    // base of this wave's B columns inside the LDS panel
    const float* bbase = Bs + 2 * hi * NC + n0 + l;

    for (int k0 = 0; k0 < K; k0 += KC) {
        // cooperative coalesced stage of B[k0 .. k0+KC) x NC
        const float4* Bg = (const float4*)(B + (size_t)k0 * NC);
        float4*       Bl = (float4*)Bs;
        constexpr int NVEC = KC * NC / 4;
        for (int i = threadIdx.x; i < NVEC; i += blockDim.x)
            Bl[i] = Bg[i];
        __syncthreads();

#pragma unroll
        for (int kk = 0; kk < KC; kk += 4) {
            v2f a = *(const v2f*)(arow + k0 + kk);     // global_load_b64
            v2f b;
            b.x = bbase[kk * NC];                      // ds_load, const offset
            b.y = bbase[(kk + 1) * NC];
            c = __builtin_amdgcn_wmma_f32_16x16x4_f32(
                    /*neg_a=*/false, a, /*neg_b=*/false, b,
                    /*c_mod=*/(short)0, c, /*reuse_a=*/false, /*reuse_b=*/false);
        }
        __syncthreads();
    }

    // C/D layout: VGPR r -> row (m0 + r + 8*hi), col (n0 + l)
    float* drow = D + (size_t)(m0 + 8 * hi) * NC + n0 + l;
#pragma unroll
    for (int r = 0; r < 8; ++r)
        drow[(size_t)r * NC] = c[r];
}

// ---------------- edge aggregation, F=128 (one wave per edge) ----------------
__global__ __launch_bounds__(256) void aggregate_f128_kernel(
    const float* __restrict__ h, const float* __restrict__ dis,
    const int* __restrict__ src, const int* __restrict__ dst,
    float* __restrict__ agg, int e) {
    int wid  = (blockIdx.x * blockDim.x + threadIdx.x) >> 5;
    int lane = threadIdx.x & 31;
    if (wid >= e) return;
    int s = src[wid], d = dst[wid];
    float w = dis[s] * dis[d];
    float4 v = ((const float4*)(h + (size_t)s * H_DIM))[lane];
    float* ad = agg + (size_t)d * H_DIM + lane * 4;
    atomicAdd(ad + 0, v.x * w);
    atomicAdd(ad + 1, v.y * w);
    atomicAdd(ad + 2, v.z * w);
    atomicAdd(ad + 3, v.w * w);
}

// ---------------- edge aggregation, F=64 (16 lanes per edge) ----------------
__global__ __launch_bounds__(256) void aggregate_f64_kernel(
    const float* __restrict__ h, const float* __restrict__ dis,
    const int* __restrict__ src, const int* __restrict__ dst,
    float* __restrict__ agg, int e) {
    int tid = blockIdx.x * blockDim.x + threadIdx.x;
    int eid = tid >> 4;
    int l   = tid & 15;
    if (eid >= e) return;
    int s = src[eid], d = dst[eid];
    float w = dis[s] * dis[d];
    float4 v = ((const float4*)(h + (size_t)s * C_DIM))[l];
    float* ad = agg + (size_t)d * C_DIM + l * 4;
    atomicAdd(ad + 0, v.x * w);
    atomicAdd(ad + 1, v.y * w);
    atomicAdd(ad + 2, v.z * w);
    atomicAdd(ad + 3, v.w * w);
}

// ---------------- finalize: agg = [relu](agg + hpre*dis^2 + bias) ----------------
__global__ __launch_bounds__(256) void finalize_kernel(
    float* __restrict__ agg, const float* __restrict__ hpre,
    const float* __restrict__ dis, const float* __restrict__ bias,
    int total, int shift, int mask, int do_relu) {
    int i = blockIdx.x * blockDim.x + threadIdx.x;
    if (i >= total) return;
    int node = i >> shift;
    int ch   = i & mask;
    float s  = dis[node];
    float v  = agg[i] + hpre[i] * (s * s) + bias[ch];
    agg[i]   = do_relu ? fmaxf(v, 0.0f) : v;
}

// ---------------- link-prediction scores (16 lanes per edge, C=64) ----------------
__global__ __launch_bounds__(256) void score_kernel(
    const float* __restrict__ h2, const int* __restrict__ pos,
    const int* __restrict__ neg, float* __restrict__ out) {
    int tid = blockIdx.x * blockDim.x + threadIdx.x;
    int eid = tid >> 4;
    int l   = tid & 15;
    if (eid >= 2 * EP_EDGES) return;
    int a, b;
    if (eid < EP_EDGES) {
        a = pos[eid];
        b = pos[EP_EDGES + eid];
    } else {
        int e2 = eid - EP_EDGES;
        a = neg[e2];
        b = neg[EP_EDGES + e2];
    }
    float4 xa = ((const float4*)(h2 + (size_t)a * C_DIM))[l];
    float4 xb = ((const float4*)(h2 + (size_t)b * C_DIM))[l];
    float p = xa.x * xb.x + xa.y * xb.y + xa.z * xb.z + xa.w * xb.w;
    // reduce over the 16-lane group (wave32)
    p += __shfl_xor(p, 1, 32);
    p += __shfl_xor(p, 2, 32);
    p += __shfl_xor(p, 4, 32);
    p += __shfl_xor(p, 8, 32);
    if (l == 0) out[eid] = p;
}

// ---------------- host driver ----------------

extern "C" void kernel_launch(void* const* d_in, const int* in_sizes, int n_in,
                              void* d_out, int out_size, void* d_ws, size_t ws_size,
                              hipStream_t stream) {
    const float* x    = (const float*)d_in[0];
    // d_in[1] = masked_nodes (unused by reference output)
    const int*   pos  = (const int*)d_in[2];   // [2, EP] row-major
    const int*   neg  = (const int*)d_in[3];   // [2, EP]
    const int*   edge = (const int*)d_in[4];   // [2, E]: src = edge[0..E), dst = edge[E..2E)
    const float* W1   = (const float*)d_in[5];
    const float* b1   = (const float*)d_in[6];
    const float* W2   = (const float*)d_in[7];
    const float* b2   = (const float*)d_in[8];
    float* out = (float*)d_out;

    const int* e_src = edge;
    const int* e_dst = edge + N_EDGES;

    // workspace layout (floats)
    float* ws   = (float*)d_ws;
    float* dis  = ws;                           // N (deg -> rsqrt), padded
    float* h1   = ws + 102400;                  // N*H  pre-activation
    float* agg1 = h1 + (size_t)N_NODES * H_DIM; // N*H  -> becomes h1_final in place
    float* h2   = h1;                           // reuse: N*C pre-activation
    float* agg2 = h1 + (size_t)N_NODES * C_DIM; // reuse: N*C -> h2_final in place

    const int BT = 256;

    // 1) degrees -> D^{-1/2}
    fill_zero_kernel<<<(N_NODES + BT - 1) / BT, BT, 0, stream>>>(dis, N_NODES);
    degree_count_kernel<<<(N_EDGES + BT - 1) / BT, BT, 0, stream>>>(e_dst, dis, N_EDGES);
    deg_to_dis_kernel<<<(N_NODES + BT - 1) / BT, BT, 0, stream>>>(dis, N_NODES);

    // 2) layer 1: h1 = x @ W1  (M=100000, K=256, NC=128; 8 waves, KC=64 -> 32 KB LDS)
    gemm_wmma_lds_kernel<64, H_DIM>
        <<<N_NODES / 16, (H_DIM / 16) * 32, 0, stream>>>(x, W1, h1, F_IN);

    int tot1 = N_NODES * H_DIM;
    fill_zero_kernel<<<(tot1 + BT - 1) / BT, BT, 0, stream>>>(agg1, tot1);
    aggregate_f128_kernel<<<(N_EDGES * 32 + BT - 1) / BT, BT, 0, stream>>>(
        h1, dis, e_src, e_dst, agg1, N_EDGES);
    finalize_kernel<<<(tot1 + BT - 1) / BT, BT, 0, stream>>>(
        agg1, h1, dis, b1, tot1, 7, H_DIM - 1, 1);   // relu; agg1 = h1_final

    // 3) layer 2: h2 = h1_final @ W2  (K=128, NC=64; 4 waves, KC=128 -> whole W2, 32 KB LDS)
    gemm_wmma_lds_kernel<128, C_DIM>
        <<<N_NODES / 16, (C_DIM / 16) * 32, 0, stream>>>(agg1, W2, h2, H_DIM);

    int tot2 = N_NODES * C_DIM;
    fill_zero_kernel<<<(tot2 + BT - 1) / BT, BT, 0, stream>>>(agg2, tot2);
    aggregate_f64_kernel<<<(N_EDGES * 16 + BT - 1) / BT, BT, 0, stream>>>(
        h2, dis, e_src, e_dst, agg2, N_EDGES);
    finalize_kernel<<<(tot2 + BT - 1) / BT, BT, 0, stream>>>(
        agg2, h2, dis, b2, tot2, 6, C_DIM - 1, 0);   // no relu; agg2 = h2_final

    // 4) scores over concat(pos, neg) edges + trailing zeros output
    score_kernel<<<(2 * EP_EDGES * 16 + BT - 1) / BT, BT, 0, stream>>>(agg2, pos, neg, out);
    fill_zero_kernel<<<(2 * 2 * EP_EDGES + BT - 1) / BT, BT, 0, stream>>>(
        out + 2 * EP_EDGES, 2 * 2 * EP_EDGES);
}